// dynamic_graph_constructor_154618823137
// MI455X (gfx1250) — compile-verified
//
#include <hip/hip_runtime.h>
#include <cstddef>
#include <cstdint>

typedef __attribute__((ext_vector_type(2))) float v2f;
typedef __attribute__((ext_vector_type(8))) float v8f;

#define NODES 1024
#define BATCH 16
#define SEQT  24
#define NLAY  24
#define NLANE (NODES * BATCH)                    // 16384 independent GRU lanes
#define ADJ_ELEMS ((size_t)NLAY * NODES * NODES) // 25,165,824 floats

// ---------------------------------------------------------------------------
// gfx1250 hardware tanh (TRANS op). sigmoid(x) = 0.5*tanh(0.5x)+0.5 gives one
// transcendental per gate instead of exp+rcp.
// ---------------------------------------------------------------------------
__device__ __forceinline__ float fast_tanh(float x) {
#if __has_builtin(__builtin_amdgcn_tanhf)
  return __builtin_amdgcn_tanhf(x);
#elif __has_builtin(__builtin_amdgcn_tanh_f32)
  return __builtin_amdgcn_tanh_f32(x);
#else
  float r;
  asm volatile("v_tanh_f32 %0, %1" : "=v"(r) : "v"(x));
  return r;
#endif
}

__device__ __forceinline__ float fast_sigmoid(float x) {
  return fmaf(0.5f, fast_tanh(0.5f * x), 0.5f);
}

// ---------------------------------------------------------------------------
// Kernel 1: systolic GRU. One wave per (node,batch) GRU lane; SIMD lane j owns
// layer j (weights in per-lane registers). At step s, lane j processes global
// time gt = s - j; its input is lane j-1's output from step s-1, moved with a
// rotate-by-1 lane permute (lane 0 reads lane 23). Because the horizon
// feedback x(gt) = out23(gt-24) is produced by lane 23 at step gt-1, all
// `horizon` passes fuse into one pipeline of 24*horizon + 23 steps:
//   - serial dependent path: 215 cells instead of 4608
//   - 16384 waves instead of 512 -> throughput-bound with full occupancy
// ---------------------------------------------------------------------------
__global__ __launch_bounds__(256) void gru_systolic_kernel(
    const float* __restrict__ xin,   // [B,1,NODES,T]
    const float* __restrict__ w_ih,  // [L,3] gates r,z,n
    const float* __restrict__ w_hh,
    const float* __restrict__ b_ih,
    const float* __restrict__ b_hh,
    const int*   __restrict__ horizon,
    float* __restrict__ hfin)        // [L][NLANE]
{
  const int lane = threadIdx.x & 31;                       // layer id (0..23 live)
  const int wid  = (blockIdx.x << 3) + (threadIdx.x >> 5); // GRU lane 0..16383
  const int node = wid >> 4;                               // wid = node*B + b
  const int b    = wid & 15;

  // Per-lane (= per-layer) weights; lanes 24..31 clamp to layer 23 (unused).
  const int   l   = (lane < NLAY) ? lane : (NLAY - 1);
  const float wi0 = w_ih[l*3+0], wi1 = w_ih[l*3+1], wi2 = w_ih[l*3+2];
  const float wh0 = w_hh[l*3+0], wh1 = w_hh[l*3+1], wh2 = w_hh[l*3+2];
  const float bc0 = b_ih[l*3+0] + b_hh[l*3+0];
  const float bc1 = b_ih[l*3+1] + b_hh[l*3+1];
  const float bi2 = b_ih[l*3+2], bh2 = b_hh[l*3+2];

  // Input sequence: lane t holds x(t); broadcast with __shfl during phase 1.
  const float* src = xin + ((size_t)b * NODES + node) * SEQT;
  const float  xv  = (lane < SEQT) ? src[lane] : 0.0f;

  const int hor  = *horizon;            // 8
  const int gmax = hor * SEQT - 1;      // last global timestep (191)
  const int send = gmax + NLAY - 1;     // last pipeline step (214)
  const int srcl = (lane == 0) ? (NLAY - 1) : (lane - 1);  // rotate-by-1 source

  float h = 0.0f;     // lane j's layer-j hidden state
  float outp = 0.0f;  // lane j's layer-j output from previous step

  // Phase 1: steps 0..23 — inject x(s) into lane 0.
  for (int s = 0; s < SEQT; ++s) {
    const float fb   = __shfl(outp, srcl, 32);
    const float xi   = __shfl(xv, s, 32);
    const float xcur = (lane == 0) ? xi : fb;
    const bool  act  = (lane <= s);
    const float hp = h;
    const float r  = fast_sigmoid(fmaf(wi0, xcur, fmaf(wh0, hp, bc0)));
    const float z  = fast_sigmoid(fmaf(wi1, xcur, fmaf(wh1, hp, bc1)));
    const float n  = fast_tanh(fmaf(wi2, xcur, bi2) + r * fmaf(wh2, hp, bh2));
    const float hn = fmaf(z, hp - n, n);           // (1-z)*n + z*h
    h = act ? hn : hp;
    outp = h;
  }

  // Phase 2: steps 24..send — lane 0's input is the horizon feedback
  // out23(s-24), produced by lane 23 one step earlier.
  #pragma unroll 2
  for (int s = SEQT; s <= send; ++s) {
    const float xcur = __shfl(outp, srcl, 32);
    const bool  act  = (lane >= s - gmax) && (lane < NLAY);
    const float hp = h;
    const float r  = fast_sigmoid(fmaf(wi0, xcur, fmaf(wh0, hp, bc0)));
    const float z  = fast_sigmoid(fmaf(wi1, xcur, fmaf(wh1, hp, bc1)));
    const float n  = fast_tanh(fmaf(wi2, xcur, bi2) + r * fmaf(wh2, hp, bh2));
    const float hn = fmaf(z, hp - n, n);
    h = act ? hn : hp;
    outp = h;
  }

  if (lane < NLAY) hfin[(size_t)lane * NLANE + wid] = h;
}

// ---------------------------------------------------------------------------
// Kernel 2: per-(layer,node) row normalization (norm clamped at eps=1e-8) and
// the h_out output ([nodes, L, B, 1] = raw hidden values, transposed).
// ---------------------------------------------------------------------------
__global__ __launch_bounds__(256) void normalize_kernel(
    const float* __restrict__ hfin,  // [L][NODES][B] contiguous
    float* __restrict__ hhat,        // [L][NODES][B] normalized rows
    float* __restrict__ hout)        // [NODES][L][B]
{
  const int idx  = blockIdx.x * 256 + threadIdx.x;   // 0..NLAY*NODES-1
  const int l    = idx >> 10;
  const int node = idx & 1023;
  const float* src = hfin + (size_t)idx * BATCH;
  float v[BATCH];
  float s = 0.0f;
  #pragma unroll
  for (int bb = 0; bb < BATCH; ++bb) { v[bb] = src[bb]; s = fmaf(v[bb], v[bb], s); }
  const float inv = 1.0f / fmaxf(sqrtf(s), 1e-8f);
  float* dh = hhat + (size_t)idx * BATCH;
  float* dq = hout + ((size_t)node * NLAY + l) * BATCH;
  #pragma unroll
  for (int bb = 0; bb < BATCH; ++bb) { dh[bb] = v[bb] * inv; dq[bb] = v[bb]; }
}

// ---------------------------------------------------------------------------
// Kernel 3: adj[l] = Hhat_l * Hhat_l^T via V_WMMA_F32_16X16X4_F32, K=16 as 4
// chained K=4 WMMAs. One wave per 16x16 output tile; 8 waves/block cover 8
// consecutive j-tiles of one i row (contiguous 16x128 store footprint).
// Output is 100 MB -> HBM-store-bound (~4.3 us floor at 23.3 TB/s); the 1.5 MB
// Hhat input stays L2-resident.
//
// A layout (ISA 32-bit A 16x4): lane L holds M=L%16, VGPR v -> K=2*(L/16)+v.
// B mirrored (lane L: N=L%16, VGPR v -> K=2*(L/16)+v), with
// B[k][n] = Hhat[j16+n][k] -> same gather as A but on the j block.
// C/D layout: VGPR r -> M = r + 8*(L/16), N = L%16.
// ---------------------------------------------------------------------------
__global__ __launch_bounds__(256) void gram_wmma_kernel(
    const float* __restrict__ hhat,  // [L][NODES][B]
    float* __restrict__ adj)         // [L][NODES][NODES]
{
  const int lane = threadIdx.x & 31;
  const int wave = threadIdx.x >> 5;
  const int tile = blockIdx.x * 8 + wave;   // l*4096 + i*64 + j
  const int l  = tile >> 12;
  const int ib = (tile >> 6) & 63;
  const int jb = tile & 63;

  const int m     = lane & 15;
  const int khalf = (lane >> 4) << 1;
  const float* pa = hhat + ((size_t)l * NODES + ib * 16 + m) * BATCH + khalf;
  const float* pb = hhat + ((size_t)l * NODES + jb * 16 + m) * BATCH + khalf;

  v8f acc = {0.f, 0.f, 0.f, 0.f, 0.f, 0.f, 0.f, 0.f};
  #pragma unroll
  for (int c = 0; c < 4; ++c) {
    const v2f av = *(const v2f*)(pa + 4 * c);   // 8B-aligned (khalf+4c even)
    const v2f bv = *(const v2f*)(pb + 4 * c);
    acc = __builtin_amdgcn_wmma_f32_16x16x4_f32(
        /*neg_a=*/false, av, /*neg_b=*/false, bv,
        /*c_mod=*/(short)0, acc, /*reuse_a=*/false, /*reuse_b=*/false);
  }

  float* out = adj + ((size_t)l * NODES + ib * 16 + ((lane >> 4) * 8)) * NODES
             + jb * 16 + (lane & 15);
  #pragma unroll
  for (int r = 0; r < 8; ++r) out[(size_t)r * NODES] = acc[r];
}

// ---------------------------------------------------------------------------
extern "C" void kernel_launch(void* const* d_in, const int* in_sizes, int n_in,
                              void* d_out, int out_size, void* d_ws, size_t ws_size,
                              hipStream_t stream) {
  (void)in_sizes; (void)n_in; (void)out_size; (void)ws_size;
  const float* xin = (const float*)d_in[0];
  const float* wih = (const float*)d_in[1];
  const float* whh = (const float*)d_in[2];
  const float* bih = (const float*)d_in[3];
  const float* bhh = (const float*)d_in[4];
  const int*   hor = (const int*)d_in[5];

  float* out  = (float*)d_out;
  float* hfin = (float*)d_ws;                       // [L][NLANE]  1.5 MB
  float* hhat = hfin + (size_t)NLAY * NLANE;        // [L][NLANE]  1.5 MB

  gru_systolic_kernel<<<NLANE / 8, 256, 0, stream>>>(xin, wih, whh, bih, bhh, hor, hfin);
  normalize_kernel<<<(NLAY * NODES) / 256, 256, 0, stream>>>(hfin, hhat, out + ADJ_ELEMS);
  gram_wmma_kernel<<<(NLAY * 64 * 64) / 8, 256, 0, stream>>>(hhat, out);
}